// MeanAggregator_41472204210463
// MI455X (gfx1250) — compile-verified
//
#include <hip/hip_runtime.h>
#include <hip/hip_bf16.h>

// Mean aggregator: out[b, :] = mean over 16 sampled neighbor feature rows.
// Pure HBM-gather-bound (~880 MB traffic -> ~38 us at 23.3 TB/s).
// Strategy: CDNA5 async-to-LDS gather (GLOBAL_LOAD_ASYNC_TO_LDS_B128,
// ASYNCcnt-tracked, in-order completion) with per-wave double buffering.
// Row-id broadcast via v_readlane (keeps the LDS pipe free for the async
// writes + ds_load_b128 accumulate reads); output stored non-temporally to
// preserve L2 for feature-row reuse (~1.6 reads/row on a 512MB table).

#define NSAMP 16
#define DFEAT 128
#define WAVES_PER_BLOCK 4
#define ROW_BYTES (DFEAT * 4)            // 512 B per feature row
#define BUF_BYTES (NSAMP * ROW_BYTES)    // 8 KB per staging buffer
#define BUF_FLOATS (NSAMP * DFEAT)       // 2048 floats

typedef __attribute__((ext_vector_type(4))) float v4f;  // native vector for NT store

__global__ __launch_bounds__(WAVES_PER_BLOCK * 32) void
MeanAggregator_41472204210463_kernel(const float* __restrict__ feats,
                                     const int* __restrict__ nidx,
                                     float* __restrict__ out,
                                     int batch)
{
    // 4 waves * 2 buffers * 8KB = 64KB static LDS per workgroup.
    __shared__ __align__(16) float lds[WAVES_PER_BLOCK * 2 * BUF_FLOATS];

    const int lane = threadIdx.x & 31;
    const int wid  = threadIdx.x >> 5;
    const int wglobal = blockIdx.x * WAVES_PER_BLOCK + wid;
    const int wstride = gridDim.x * WAVES_PER_BLOCK;

    // LDS byte address = low 32 bits of the generic pointer (aperture rule).
    const uint32_t lds_base  = (uint32_t)(uintptr_t)(&lds[0]);
    const uint32_t mybuf0    = lds_base + (uint32_t)(wid * 2) * BUF_BYTES;
    const uint64_t feat_base = (uint64_t)(uintptr_t)feats;
    float* const   myptr0    = &lds[wid * 2 * BUF_FLOATS];

    int node = wglobal;
    if (node >= batch) return;   // wave-uniform exit

    // Issue 16 async row-gathers (one full 512B row per instruction:
    // 32 lanes x 16B) for node `n` into buffer b (0/1). Tracked by ASYNCcnt.
    auto issue = [&](int n, int b) {
        int myidx = nidx[n * NSAMP + (lane & (NSAMP - 1))];
        const uint32_t dst0 = mybuf0 + (uint32_t)b * BUF_BYTES + ((uint32_t)lane << 4);
#pragma unroll
        for (int s = 0; s < NSAMP; ++s) {
            // Row id is wave-uniform: broadcast through an SGPR (VALU/SALU),
            // not ds_bpermute, to keep the LDS pipe free.
            uint32_t row  = (uint32_t)__builtin_amdgcn_readlane(myidx, s);
            uint32_t goff = (row << 9) + ((uint32_t)lane << 4); // row*512 + lane*16
            uint32_t dst  = dst0 + (uint32_t)(s * ROW_BYTES);
            // GVS mode: mem_addr = SADDR(64b features base) + VADDR(i32 byte off)
            asm volatile("global_load_async_to_lds_b128 %0, %1, %2"
                         :
                         : "v"(dst), "v"(goff), "s"(feat_base)
                         : "memory");
        }
    };

    issue(node, 0);
    int cur = 0;
    for (;;) {
        const int nxt = node + wstride;
        const bool more = (nxt < batch);
        if (more) {
            issue(nxt, cur ^ 1);                                  // prefetch next node
            // In-order async completion: cnt<=16 (the 16 just-issued prefetch
            // ops) => all rows of the current buffer have landed in LDS.
            asm volatile("s_wait_asynccnt 16" ::: "memory");
        } else {
            asm volatile("s_wait_asynccnt 0" ::: "memory");
        }

        // Accumulate current buffer: lane owns features [4*lane, 4*lane+3].
        const float* buf = myptr0 + cur * BUF_FLOATS;
        float ax = 0.f, ay = 0.f, az = 0.f, aw = 0.f;
#pragma unroll
        for (int s = 0; s < NSAMP; ++s) {
            const v4f v = *(const v4f*)(buf + s * DFEAT + lane * 4);
            ax += v.x; ay += v.y; az += v.z; aw += v.w;
        }
        v4f r;
        r.x = ax * (1.0f / NSAMP);
        r.y = ay * (1.0f / NSAMP);
        r.z = az * (1.0f / NSAMP);
        r.w = aw * (1.0f / NSAMP);
        // Write-once output: non-temporal store, don't evict feature lines.
        __builtin_nontemporal_store(r, (v4f*)(out + (size_t)node * DFEAT + lane * 4));

        if (!more) break;
        node = nxt;
        cur ^= 1;
    }
}

extern "C" void kernel_launch(void* const* d_in, const int* in_sizes, int n_in,
                              void* d_out, int out_size, void* d_ws, size_t ws_size,
                              hipStream_t stream) {
    (void)n_in; (void)d_ws; (void)ws_size; (void)out_size;
    const float* feats = (const float*)d_in[0];
    const int*   nidx  = (const int*)d_in[1];   // harness canonicalizes ints to int32
    float*       out   = (float*)d_out;

    const int batch = in_sizes[1] / NSAMP;      // 100,000 for the reference setup

    int blocks = (batch + WAVES_PER_BLOCK - 1) / WAVES_PER_BLOCK;
    if (blocks > 2048) blocks = 2048;           // ~8192 waves, ~12 nodes/wave pipeline
    if (blocks < 1) blocks = 1;

    MeanAggregator_41472204210463_kernel<<<blocks, WAVES_PER_BLOCK * 32, 0, stream>>>(
        feats, nidx, out, batch);
}